// LearnedPosMapT_87986700026183
// MI455X (gfx1250) — compile-verified
//
#include <hip/hip_runtime.h>

typedef __attribute__((ext_vector_type(2))) float v2f;
typedef __attribute__((ext_vector_type(8))) float v8f;

constexpr int WIN    = 8;
constexpr int DIM    = 384;
constexpr int GAMMA  = 8;
constexpr int NWIN   = 3136;
constexpr int BATCH  = 8;
constexpr int SLAB   = WIN * DIM;          // 3072 floats (12 KB) per window
constexpr int NWINDOWS = BATCH * NWIN;     // 25088
constexpr int WAVES_PER_BLOCK = 4;
constexpr int BLOCK  = 32 * WAVES_PER_BLOCK;
constexpr int NBLOCKS = NWINDOWS / WAVES_PER_BLOCK;  // 6272 (exact)

// CDNA5 async global->LDS staging if the toolchain exposes the builtins,
// else fall back to b128 load + ds_store (identical data movement).
#if defined(__has_builtin)
#  if __has_builtin(__builtin_amdgcn_global_load_async_to_lds_b128) && \
      __has_builtin(__builtin_amdgcn_s_wait_asynccnt)
#    define USE_ASYNC_LDS 1
#  endif
#endif
#ifndef USE_ASYNC_LDS
#  define USE_ASYNC_LDS 0
#endif

#if USE_ASYNC_LDS
// Builtin signature (from hipcc diagnostic): param0 = global (AS1) int4*,
// param1 = LDS (AS3) int4*, then imm offset, imm cpol.
typedef int b128_t __attribute__((vector_size(16)));
typedef __attribute__((address_space(1))) b128_t b128_g;
typedef __attribute__((address_space(3))) b128_t b128_l;
#endif

__global__ __launch_bounds__(BLOCK)
void LearnedPosMapT_kernel(const float* __restrict__ x,
                           const float* __restrict__ table,  // (2*WIN-1, GAMMA) = 15x8
                           const float* __restrict__ bias,   // (1, WIN, 1) = 8
                           float* __restrict__ out)
{
  __shared__ float sTab[(2 * WIN - 1) * GAMMA];      // 120 floats
  __shared__ float sSlab[WAVES_PER_BLOCK][SLAB];     // 4 x 12 KB

  if (threadIdx.x < (2 * WIN - 1) * GAMMA) sTab[threadIdx.x] = table[threadIdx.x];
  __syncthreads();

  const int warp = threadIdx.x >> 5;
  const int lane = threadIdx.x & 31;
  const int win  = blockIdx.x * WAVES_PER_BLOCK + warp;
  if (win >= NWINDOWS) return;  // wave-uniform (grid is exact)

  const float* __restrict__ px = x + (size_t)win * SLAB;
  float* __restrict__       po = out + (size_t)win * SLAB;
  float* __restrict__       sl = sSlab[warp];

  __builtin_prefetch(px, 0, 1);  // global_prefetch_b8

  // Stage the 8x384 contiguous window slab into LDS (24 x 512B per wave).
#if USE_ASYNC_LDS
  #pragma unroll
  for (int i = 0; i < SLAB / 128; ++i) {
    const int e = i * 128 + lane * 4;
    __builtin_amdgcn_global_load_async_to_lds_b128(
        (b128_g*)(px + e), (b128_l*)(sl + e),
        /*imm offset=*/0, /*cpol=*/0);
  }
  __builtin_amdgcn_s_wait_asynccnt(0);  // slab visible before DS reads below
#else
  #pragma unroll
  for (int i = 0; i < SLAB / 128; ++i) {
    const int e = i * 128 + lane * 4;
    const float4 f = *reinterpret_cast<const float4*>(px + e);
    *reinterpret_cast<float4*>(sl + e) = f;
  }
#endif

  // Per-output-row bias (uniform across lanes -> scalar loads).
  float br[WIN];
  #pragma unroll
  for (int r = 0; r < WIN; ++r) br[r] = bias[r];

  // Lane -> fragment coordinates (ISA 7.12.2 layouts, wave32):
  //   A (16x4): lane holds row M = lane%16, K pair {0,1} (lanes 0-15) or {2,3}.
  //   B (4x16) / D (16x16): column N = lane%16, K/M split across lane halves.
  const int mcol = lane & 15;         // A row (m) and B/D column index in tile
  const int k0   = (lane >> 4) << 1;  // K base within a 4-wide step: 0 or 2

  // Process s-channels in adjacent pairs: adjacent s are adjacent in memory
  // (d = v*8 + s), so LDS reads and global stores become 8-byte ops.
  for (int sp = 0; sp < GAMMA / 2; ++sp) {
    const int s0 = 2 * sp;

    // A[m,k] = table[(m - n + 7)*GAMMA + s]; zero rows for padded m >= 8.
    v2f A0a = {0.f, 0.f}, A1a = {0.f, 0.f};  // s = s0
    v2f A0b = {0.f, 0.f}, A1b = {0.f, 0.f};  // s = s0 + 1
    if (mcol < WIN) {
      const float2 w0 = *reinterpret_cast<const float2*>(&sTab[(mcol - (k0 + 0) + 7) * GAMMA + s0]);
      const float2 w1 = *reinterpret_cast<const float2*>(&sTab[(mcol - (k0 + 1) + 7) * GAMMA + s0]);
      const float2 w4 = *reinterpret_cast<const float2*>(&sTab[(mcol - (k0 + 4) + 7) * GAMMA + s0]);
      const float2 w5 = *reinterpret_cast<const float2*>(&sTab[(mcol - (k0 + 5) + 7) * GAMMA + s0]);
      A0a.x = w0.x; A0b.x = w0.y;
      A0a.y = w1.x; A0b.y = w1.y;
      A1a.x = w4.x; A1b.x = w4.y;
      A1a.y = w5.x; A1b.y = w5.y;
    }

    #pragma unroll
    for (int vt = 0; vt < 3; ++vt) {
      const int coloff = (vt * 16 + mcol) * GAMMA + s0;  // v*8 + s0, 8B aligned
      // B[k, col] = x[n=k, v=col, s], s-pair per b64 LDS read.
      const float2 x0 = *reinterpret_cast<const float2*>(&sl[(k0 + 0) * DIM + coloff]);
      const float2 x1 = *reinterpret_cast<const float2*>(&sl[(k0 + 1) * DIM + coloff]);
      const float2 x4 = *reinterpret_cast<const float2*>(&sl[(k0 + 4) * DIM + coloff]);
      const float2 x5 = *reinterpret_cast<const float2*>(&sl[(k0 + 5) * DIM + coloff]);
      v2f B0a = {x0.x, x1.x}, B1a = {x4.x, x5.x};
      v2f B0b = {x0.y, x1.y}, B1b = {x4.y, x5.y};

      v8f acca = {0.f, 0.f, 0.f, 0.f, 0.f, 0.f, 0.f, 0.f};
      v8f accb = {0.f, 0.f, 0.f, 0.f, 0.f, 0.f, 0.f, 0.f};
      // D = A(16x4) x B(4x16) + C, chained over K = n in [0,4) then [4,8).
      acca = __builtin_amdgcn_wmma_f32_16x16x4_f32(false, A0a, false, B0a,
                                                   (short)0, acca, false, false);
      acca = __builtin_amdgcn_wmma_f32_16x16x4_f32(false, A1a, false, B1a,
                                                   (short)0, acca, false, false);
      accb = __builtin_amdgcn_wmma_f32_16x16x4_f32(false, A0b, false, B0b,
                                                   (short)0, accb, false, false);
      accb = __builtin_amdgcn_wmma_f32_16x16x4_f32(false, A1b, false, B1b,
                                                   (short)0, accb, false, false);

      // D rows 0..7 are the real outputs: lane<16 holds (M=r, N=lane).
      // Store the s-pair as one b64 per row.
      if (lane < 16) {
        #pragma unroll
        for (int r = 0; r < WIN; ++r) {
          float2 o;
          o.x = acca[r] + br[r];
          o.y = accb[r] + br[r];
          *reinterpret_cast<float2*>(&po[r * DIM + coloff]) = o;
        }
      }
    }
  }
}

extern "C" void kernel_launch(void* const* d_in, const int* in_sizes, int n_in,
                              void* d_out, int out_size, void* d_ws, size_t ws_size,
                              hipStream_t stream) {
  const float* x     = (const float*)d_in[0];  // (8, 3136, 8, 384) f32
  const float* table = (const float*)d_in[1];  // (15, 8) f32
  const float* bias  = (const float*)d_in[2];  // (1, 8, 1) f32
  float* out         = (float*)d_out;          // (8, 3136, 8, 384) f32

  LearnedPosMapT_kernel<<<NBLOCKS, BLOCK, 0, stream>>>(x, table, bias, out);
}